// ContamRiskGNN_39994735460486
// MI455X (gfx1250) — compile-verified
//
#include <hip/hip_runtime.h>
#include <hip/hip_bf16.h>
#include <math.h>

#define NN 50000
#define EE 800000
#define BB 64
#define HH 128
#define HEADS 4
#define DD 32
#define NODE_DIM 48
#define EDGE_DIM 7
#define NLAYER 3
#define RCLS 4

typedef float v2f __attribute__((ext_vector_type(2)));
typedef float v8f __attribute__((ext_vector_type(8)));

__device__ __forceinline__ float sigm(float x){ return 1.0f/(1.0f+expf(-x)); }
__device__ __forceinline__ float silu(float x){ return x/(1.0f+expf(-x)); }

// order-preserving float<->uint mapping for atomicMax on floats
__device__ __forceinline__ unsigned flipf(float f){
  unsigned u = __float_as_uint(f);
  return (u & 0x80000000u) ? ~u : (u | 0x80000000u);
}
__device__ __forceinline__ float unflipf(unsigned u){
  return __uint_as_float((u & 0x80000000u) ? (u ^ 0x80000000u) : ~u);
}

// ---------------------------------------------------------------------------
// C[M x NC] = act(A[M x K] @ W[K x NC] + bias), f32 WMMA 16x16x4.
// grid = (M/16, NC/16), block = 32 (one wave per 16x16 tile). M%16==0, K%4==0,
// NC%16==0. act: 0 = none, 1 = SiLU.
// A-frag (doc layout): lane<16 -> M=lane, K=k,k+1 ; lane>=16 -> M=lane-16, K=k+2,k+3
// B-frag (symmetric):  lane<16 -> N=lane, K=k,k+1 ; lane>=16 -> N=lane-16, K=k+2,k+3
// D (doc layout): reg r, lane l -> M = r + 8*(l>>4), N = l&15
// ---------------------------------------------------------------------------
__global__ __launch_bounds__(32)
void gemm_wmma_k(const float* __restrict__ A, const float* __restrict__ W,
                 const float* __restrict__ bias, float* __restrict__ C,
                 int M, int K, int NC, int act)
{
  const int lane = threadIdx.x;
  const int mh   = lane & 15;
  const int half = lane >> 4;
  const int row0 = blockIdx.x << 4;
  const int col0 = blockIdx.y << 4;
  const float* Arow = A + (size_t)(row0 + mh) * K;
  const float* Wc   = W + (col0 + mh);
  v8f acc = {0.f,0.f,0.f,0.f,0.f,0.f,0.f,0.f};
#if __has_builtin(__builtin_amdgcn_wmma_f32_16x16x4_f32)
  for (int k = 0; k < K; k += 4) {
    v2f a, b;
    a[0] = Arow[k + 2*half];
    a[1] = Arow[k + 2*half + 1];
    b[0] = Wc[(size_t)(k + 2*half) * NC];
    b[1] = Wc[(size_t)(k + 2*half + 1) * NC];
    acc = __builtin_amdgcn_wmma_f32_16x16x4_f32(false, a, false, b,
                                                (short)0, acc, false, false);
  }
#else
  for (int k = 0; k < K; ++k) {
    float w = W[(size_t)k * NC + col0 + mh];
    #pragma unroll
    for (int r = 0; r < 8; ++r)
      acc[r] += A[(size_t)(row0 + r + 8*half) * K + k] * w;
  }
#endif
  float bv = bias ? bias[col0 + mh] : 0.0f;
  #pragma unroll
  for (int r = 0; r < 8; ++r) {
    float v = acc[r] + bv;
    if (act == 1) v = silu(v);
    C[(size_t)(row0 + r + 8*half) * NC + col0 + mh] = v;
  }
}

// ---------------------------------------------------------------------------
// Per-node: v = in[+convbias][*gate-mean] ; LayerNorm(128) ; SiLU ; h = v(+h)
// grid = NN, block = 128 (one node per block)
// ---------------------------------------------------------------------------
__global__ __launch_bounds__(128)
void node_ln_silu_k(const float* __restrict__ in, const float* __restrict__ convbias,
                    const float* __restrict__ gsum, const float* __restrict__ deg,
                    const float* __restrict__ g, const float* __restrict__ b,
                    float* __restrict__ hout, int res)
{
  __shared__ float red[128];
  const int n = blockIdx.x;
  const int ch = threadIdx.x;
  float v = in[(size_t)n * HH + ch];
  if (convbias) v += convbias[ch];
  if (gsum) {
    float dg = deg[n]; dg = dg < 1.0f ? 1.0f : dg;
    v *= gsum[n] / dg;
  }
  red[ch] = v; __syncthreads();
  for (int s = 64; s > 0; s >>= 1) { if (ch < s) red[ch] += red[ch + s]; __syncthreads(); }
  float m = red[0] * (1.0f / HH); __syncthreads();
  float d0 = v - m;
  red[ch] = d0 * d0; __syncthreads();
  for (int s = 64; s > 0; s >>= 1) { if (ch < s) red[ch] += red[ch + s]; __syncthreads(); }
  float var = red[0] * (1.0f / HH);
  float y = d0 * rsqrtf(var + 1e-5f) * g[ch] + b[ch];
  y = silu(y);
  size_t idx = (size_t)n * HH + ch;
  hout[idx] = res ? (hout[idx] + y) : y;
}

// generic row LayerNorm (no activation), dim arbitrary; block=128, grid=rows
__global__ __launch_bounds__(128)
void ln_rows_k(const float* __restrict__ in, const float* __restrict__ g,
               const float* __restrict__ b, float* __restrict__ out, int dim)
{
  __shared__ float red[128];
  const int row = blockIdx.x, tid = threadIdx.x;
  const float* rp = in + (size_t)row * dim;
  float s = 0.0f;
  for (int j = tid; j < dim; j += 128) s += rp[j];
  red[tid] = s; __syncthreads();
  for (int k = 64; k > 0; k >>= 1) { if (tid < k) red[tid] += red[tid + k]; __syncthreads(); }
  float m = red[0] / (float)dim; __syncthreads();
  float vv = 0.0f;
  for (int j = tid; j < dim; j += 128) { float d = rp[j] - m; vv += d * d; }
  red[tid] = vv; __syncthreads();
  for (int k = 64; k > 0; k >>= 1) { if (tid < k) red[tid] += red[tid + k]; __syncthreads(); }
  float var = red[0] / (float)dim;
  float rs = rsqrtf(var + 1e-5f);
  for (int j = tid; j < dim; j += 128)
    out[(size_t)row * dim + j] = (rp[j] - m) * rs * g[j] + b[j];
}

// ---------------------------------------------------------------------------
// Per edge (one wave): attention logits per head + edge-gate MLP.
// channel = head*32 + lane -> per-head dot is a 32-lane shfl reduction.
// block = 256 (8 waves/edges per block)
// ---------------------------------------------------------------------------
__global__ __launch_bounds__(256)
void edge_logits_k(const int* __restrict__ ei, const float* __restrict__ ea,
                   const float* __restrict__ xl, const float* __restrict__ xr,
                   const float* __restrict__ We, const float* __restrict__ att,
                   const float* __restrict__ egW1, const float* __restrict__ egb1,
                   const float* __restrict__ egW2, const float* __restrict__ egb2,
                   float* __restrict__ logits, float* __restrict__ gate)
{
  const int wid  = threadIdx.x >> 5;
  const int lane = threadIdx.x & 31;
  const int e = blockIdx.x * 8 + wid;
  if (e >= EE) return;
  const int s = ei[e];
  const int d = ei[EE + e];
  float eav[EDGE_DIM];
  #pragma unroll
  for (int j = 0; j < EDGE_DIM; ++j) eav[j] = ea[(size_t)e * EDGE_DIM + j];

  float part[HEADS];
  #pragma unroll
  for (int h = 0; h < HEADS; ++h) {
    int ch = h * 32 + lane;
    float ev = 0.0f;
    #pragma unroll
    for (int j = 0; j < EDGE_DIM; ++j) ev += eav[j] * We[j * HH + ch];
    float z = xl[(size_t)s * HH + ch] + xr[(size_t)d * HH + ch] + ev;
    z = z > 0.0f ? z : 0.2f * z;           // leaky_relu(0.2)
    part[h] = z * att[ch];
  }
  // edge-gate: 7 -> 14 (silu) -> 1 (sigmoid); lanes 0..13 each own a hidden unit
  float gp = 0.0f;
  if (lane < 14) {
    float hs = egb1[lane];
    #pragma unroll
    for (int j = 0; j < EDGE_DIM; ++j) hs += eav[j] * egW1[j * 14 + lane];
    gp = silu(hs) * egW2[lane];
  }
  #pragma unroll
  for (int off = 16; off > 0; off >>= 1) {
    #pragma unroll
    for (int h = 0; h < HEADS; ++h) part[h] += __shfl_xor(part[h], off, 32);
    gp += __shfl_xor(gp, off, 32);
  }
  if (lane == 0) {
    #pragma unroll
    for (int h = 0; h < HEADS; ++h) logits[(size_t)e * HEADS + h] = part[h];
    gate[e] = sigm(gp + egb2[0]);
  }
}

// segment max of logits over dst via flipped-uint atomicMax; thread per edge
__global__ __launch_bounds__(256)
void edge_max_k(const int* __restrict__ ei, const float* __restrict__ logits,
                unsigned* __restrict__ mxk)
{
  const int e = blockIdx.x * 256 + threadIdx.x;
  if (e >= EE) return;
  const int d = ei[EE + e];
  #pragma unroll
  for (int h = 0; h < HEADS; ++h)
    atomicMax(&mxk[(size_t)d * HEADS + h], flipf(logits[(size_t)e * HEADS + h]));
}

// a = exp(logit - mx[dst]) stored in-place; ssum += a; gate-sum & degree
__global__ __launch_bounds__(256)
void edge_expsum_k(const int* __restrict__ ei, float* __restrict__ logits,
                   const unsigned* __restrict__ mxk, const float* __restrict__ gate,
                   float* __restrict__ ssum, float* __restrict__ gs,
                   float* __restrict__ deg)
{
  const int e = blockIdx.x * 256 + threadIdx.x;
  if (e >= EE) return;
  const int d = ei[EE + e];
  #pragma unroll
  for (int h = 0; h < HEADS; ++h) {
    float a = expf(logits[(size_t)e * HEADS + h] - unflipf(mxk[(size_t)d * HEADS + h]));
    logits[(size_t)e * HEADS + h] = a;
    atomicAdd(&ssum[(size_t)d * HEADS + h], a);
  }
  atomicAdd(&gs[d], gate[e]);
  atomicAdd(&deg[d], 1.0f);
}

// out[dst] += xl[src] * alpha  (one wave per edge, channel = head*32+lane)
__global__ __launch_bounds__(256)
void edge_scatter_k(const int* __restrict__ ei, const float* __restrict__ aexp,
                    const float* __restrict__ ssum, const float* __restrict__ xl,
                    float* __restrict__ out)
{
  const int wid  = threadIdx.x >> 5;
  const int lane = threadIdx.x & 31;
  const int e = blockIdx.x * 8 + wid;
  if (e >= EE) return;
  const int s = ei[e];
  const int d = ei[EE + e];
  float alpha[HEADS];
  #pragma unroll
  for (int h = 0; h < HEADS; ++h)
    alpha[h] = aexp[(size_t)e * HEADS + h] / (ssum[(size_t)d * HEADS + h] + 1e-16f);
  #pragma unroll
  for (int h = 0; h < HEADS; ++h) {
    int ch = h * 32 + lane;
    atomicAdd(&out[(size_t)d * HH + ch], xl[(size_t)s * HH + ch] * alpha[h]);
  }
}

// item head tail: base = sigmoid(t1 @ W2 + b2); + tier bump; clamp
__global__ __launch_bounds__(256)
void item_k(const float* __restrict__ t1, const float* __restrict__ W2,
            const float* __restrict__ b2, const float* __restrict__ tier,
            const float* __restrict__ x, float* __restrict__ item_risk)
{
  const int n = blockIdx.x * 256 + threadIdx.x;
  if (n >= NN) return;
  float acc = b2[0];
  for (int j = 0; j < 64; ++j) acc += t1[(size_t)n * 64 + j] * W2[j];
  float base = sigm(acc);
  int id = 0;
  float bm = x[(size_t)n * NODE_DIM + 38];
  #pragma unroll
  for (int j = 1; j < 4; ++j) {
    float v = x[(size_t)n * NODE_DIM + 38 + j];
    if (v > bm) { bm = v; id = j; }
  }
  float r = base + sigm(tier[id]) * 0.3f;
  item_risk[n] = fminf(fmaxf(r, 0.0f), 1.0f);
}

// batch pooling: sum, max (flip-key), count ; thread per (node,channel)
__global__ __launch_bounds__(256)
void pool_k(const float* __restrict__ h, const int* __restrict__ batch,
            float* __restrict__ gsumb, unsigned* __restrict__ gmaxk,
            float* __restrict__ cnt)
{
  const long long idx = (long long)blockIdx.x * 256 + threadIdx.x;
  if (idx >= (long long)NN * HH) return;
  const int n = (int)(idx >> 7);
  const int ch = (int)(idx & 127);
  const int bn = batch[n];
  float v = h[idx];
  atomicAdd(&gsumb[(size_t)bn * HH + ch], v);
  atomicMax(&gmaxk[(size_t)bn * HH + ch], flipf(v));
  if (ch == 0) atomicAdd(&cnt[bn], 1.0f);
}

// g = concat(sum, mean, max)  -> (BB x 384)
__global__ __launch_bounds__(128)
void bin_cat_k(const float* __restrict__ gsumb, const unsigned* __restrict__ gmaxk,
               const float* __restrict__ cnt, float* __restrict__ g)
{
  const int bi = blockIdx.x, ch = threadIdx.x;
  float s = gsumb[(size_t)bi * HH + ch];
  float c = cnt[bi];
  float cc = c < 1.0f ? 1.0f : c;
  float mx = (c > 0.0f) ? unflipf(gmaxk[(size_t)bi * HH + ch]) : -INFINITY;
  g[(size_t)bi * 384 + ch]       = s;
  g[(size_t)bi * 384 + 128 + ch] = s / cc;
  g[(size_t)bi * 384 + 256 + ch] = mx;
}

// bin head tail: bin_risk = sigmoid(hh2@sW+sb); risk_logits = hh2@cW+cb
__global__ __launch_bounds__(64)
void bin_out_k(const float* __restrict__ hh2, const float* __restrict__ sW,
               const float* __restrict__ sb, const float* __restrict__ cW,
               const float* __restrict__ cb, float* __restrict__ bin_risk,
               float* __restrict__ risk_logits)
{
  const int bi = threadIdx.x;
  if (bi >= BB) return;
  float sacc = sb[0];
  float cacc[RCLS] = {cb[0], cb[1], cb[2], cb[3]};
  for (int j = 0; j < 64; ++j) {
    float v = hh2[(size_t)bi * 64 + j];
    sacc += v * sW[j];
    #pragma unroll
    for (int c = 0; c < RCLS; ++c) cacc[c] += v * cW[j * RCLS + c];
  }
  bin_risk[bi] = sigm(sacc);
  #pragma unroll
  for (int c = 0; c < RCLS; ++c) risk_logits[(size_t)bi * RCLS + c] = cacc[c];
}

// ---------------------------------------------------------------------------
extern "C" void kernel_launch(void* const* d_in, const int* in_sizes, int n_in,
                              void* d_out, int out_size, void* d_ws, size_t ws_size,
                              hipStream_t stream)
{
  (void)in_sizes; (void)n_in; (void)out_size; (void)ws_size;
  const float* x     = (const float*)d_in[0];
  const int*   ei    = (const int*)d_in[1];
  const float* ea    = (const float*)d_in[2];
  const int*   batch = (const int*)d_in[3];
  auto F = [&](int i) { return (const float*)d_in[i]; };
  // params flattened jax-pytree style (dict keys sorted)
  const float *bin_W1 = F(4), *bin_W2 = F(5), *bin_b1 = F(6), *bin_b2 = F(7),
              *bin_cW = F(8), *bin_cb = F(9), *bin_ln_b = F(10), *bin_ln_g = F(11),
              *bin_sW = F(12), *bin_sb = F(13);
  const float *in_W = F(14), *in_b = F(15), *in_beta = F(16), *in_g = F(17);
  const float *it_W1 = F(18), *it_W2 = F(19), *it_b1 = F(20), *it_b2 = F(21),
              *it_tier = F(22);
  struct Lp { const float *We,*Wl,*Wr,*att,*bias,*bl,*br,*egW1,*egW2,*egb1,*egb2,*ln_b,*ln_g; };
  Lp lp[NLAYER];
  for (int i = 0; i < NLAYER; ++i) {
    int b0 = 23 + 13 * i;
    lp[i] = { F(b0), F(b0+1), F(b0+2), F(b0+3), F(b0+4), F(b0+5), F(b0+6),
              F(b0+7), F(b0+8), F(b0+9), F(b0+10), F(b0+11), F(b0+12) };
  }

  float* out = (float*)d_out;
  float* item_risk   = out;
  float* bin_risk    = out + NN;
  float* risk_logits = out + NN + BB;
  float* hbuf        = out + NN + BB + BB * RCLS;   // h lives in its output slot

  float* ws = (float*)d_ws;
  size_t off = 0;
  float* xl      = ws + off; off += (size_t)NN * HH;
  float* xr      = ws + off; off += (size_t)NN * HH;
  float* convout = ws + off; off += (size_t)NN * HH;
  float* logits  = ws + off; off += (size_t)EE * HEADS;
  float* gate    = ws + off; off += (size_t)EE;
  unsigned* mxk  = (unsigned*)(ws + off); off += (size_t)NN * HEADS;
  float* ssum    = ws + off; off += (size_t)NN * HEADS;
  float* gs      = ws + off; off += (size_t)NN;
  float* deg     = ws + off; off += (size_t)NN;
  float* gsumb   = ws + off; off += (size_t)BB * HH;
  unsigned* gmaxk= (unsigned*)(ws + off); off += (size_t)BB * HH;
  float* cnt     = ws + off; off += (size_t)BB;
  float* gcat    = ws + off; off += (size_t)BB * 384;
  float* glnout  = ws + off; off += (size_t)BB * 384;
  float* hh1     = ws + off; off += (size_t)BB * HH;
  float* hh2     = ws + off; off += (size_t)BB * 64;
  float* t1 = xl;  // reuse after GNN layers are done

  // ---- input projection: silu(ln(x @ in_W + in_b)) ----
  gemm_wmma_k<<<dim3(NN/16, HH/16), 32, 0, stream>>>(x, in_W, in_b, convout,
                                                     NN, NODE_DIM, HH, 0);
  node_ln_silu_k<<<NN, 128, 0, stream>>>(convout, nullptr, nullptr, nullptr,
                                         in_g, in_beta, hbuf, 0);

  // ---- GATv2 layers ----
  for (int i = 0; i < NLAYER; ++i) {
    gemm_wmma_k<<<dim3(NN/16, HH/16), 32, 0, stream>>>(hbuf, lp[i].Wl, lp[i].bl,
                                                       xl, NN, HH, HH, 0);
    gemm_wmma_k<<<dim3(NN/16, HH/16), 32, 0, stream>>>(hbuf, lp[i].Wr, lp[i].br,
                                                       xr, NN, HH, HH, 0);
    edge_logits_k<<<(EE + 7) / 8, 256, 0, stream>>>(ei, ea, xl, xr, lp[i].We,
        lp[i].att, lp[i].egW1, lp[i].egb1, lp[i].egW2, lp[i].egb2, logits, gate);
    hipMemsetAsync(mxk,  0, sizeof(unsigned) * (size_t)NN * HEADS, stream);
    hipMemsetAsync(ssum, 0, sizeof(float) * (size_t)NN * HEADS, stream);
    hipMemsetAsync(gs,   0, sizeof(float) * (size_t)NN, stream);
    hipMemsetAsync(deg,  0, sizeof(float) * (size_t)NN, stream);
    hipMemsetAsync(convout, 0, sizeof(float) * (size_t)NN * HH, stream);
    edge_max_k<<<(EE + 255) / 256, 256, 0, stream>>>(ei, logits, mxk);
    edge_expsum_k<<<(EE + 255) / 256, 256, 0, stream>>>(ei, logits, mxk, gate,
                                                        ssum, gs, deg);
    edge_scatter_k<<<(EE + 7) / 8, 256, 0, stream>>>(ei, logits, ssum, xl, convout);
    node_ln_silu_k<<<NN, 128, 0, stream>>>(convout, lp[i].bias, gs, deg,
                                           lp[i].ln_g, lp[i].ln_b, hbuf, i > 0 ? 1 : 0);
  }

  // ---- item risk head ----
  gemm_wmma_k<<<dim3(NN/16, 64/16), 32, 0, stream>>>(hbuf, it_W1, it_b1, t1,
                                                     NN, HH, 64, 1);
  item_k<<<(NN + 255) / 256, 256, 0, stream>>>(t1, it_W2, it_b2, it_tier, x, item_risk);

  // ---- bin risk head ----
  hipMemsetAsync(gsumb, 0, sizeof(float) * (size_t)BB * HH, stream);
  hipMemsetAsync(gmaxk, 0, sizeof(unsigned) * (size_t)BB * HH, stream);
  hipMemsetAsync(cnt,   0, sizeof(float) * (size_t)BB, stream);
  pool_k<<<(int)(((long long)NN * HH + 255) / 256), 256, 0, stream>>>(hbuf, batch,
                                                                      gsumb, gmaxk, cnt);
  bin_cat_k<<<BB, 128, 0, stream>>>(gsumb, gmaxk, cnt, gcat);
  ln_rows_k<<<BB, 128, 0, stream>>>(gcat, bin_ln_g, bin_ln_b, glnout, 3 * HH);
  gemm_wmma_k<<<dim3(BB/16, HH/16), 32, 0, stream>>>(glnout, bin_W1, bin_b1, hh1,
                                                     BB, 3 * HH, HH, 1);
  gemm_wmma_k<<<dim3(BB/16, 64/16), 32, 0, stream>>>(hh1, bin_W2, bin_b2, hh2,
                                                     BB, HH, 64, 1);
  bin_out_k<<<1, 64, 0, stream>>>(hh2, bin_sW, bin_sb, bin_cW, bin_cb,
                                  bin_risk, risk_logits);
}